// TransInvLoss_18116172054727
// MI455X (gfx1250) — compile-verified
//
#include <hip/hip_runtime.h>
#include <math.h>
#include <stdint.h>

#define T_DIM   2048
#define B_DIM   256
#define V_DIM   90
#define P_DIM   89
#define ROWS    (T_DIM * B_DIM)
#define IGNORE_IDX 999
#define EPS_F   1e-28f
#define NTHREADS 256
#define WAVES_PER_BLK (NTHREADS / 32)
#define MAX_BLOCKS 2048

// Wave32 butterfly reductions via ds_swizzle_b32 group-of-32 XOR patterns
// (CDNA5 ISA: offset = {xor[14:10], or[9:5], and[4:0]}).
// One ds op + one VALU op per step; EXEC must be full at call sites.
__device__ __forceinline__ float swz(float v, const int imm_is_inlined_below) { return v; }

__device__ __forceinline__ float wave_sum_f32(float v) {
    v += __int_as_float(__builtin_amdgcn_ds_swizzle(__float_as_int(v), 0x401f)); // SWAPX16
    v += __int_as_float(__builtin_amdgcn_ds_swizzle(__float_as_int(v), 0x201f)); // SWAPX8
    v += __int_as_float(__builtin_amdgcn_ds_swizzle(__float_as_int(v), 0x101f)); // SWAPX4
    v += __int_as_float(__builtin_amdgcn_ds_swizzle(__float_as_int(v), 0x081f)); // SWAPX2
    v += __int_as_float(__builtin_amdgcn_ds_swizzle(__float_as_int(v), 0x041f)); // SWAPX1
    return v;
}

__device__ __forceinline__ float wave_max_f32(float v) {
    v = fmaxf(v, __int_as_float(__builtin_amdgcn_ds_swizzle(__float_as_int(v), 0x401f)));
    v = fmaxf(v, __int_as_float(__builtin_amdgcn_ds_swizzle(__float_as_int(v), 0x201f)));
    v = fmaxf(v, __int_as_float(__builtin_amdgcn_ds_swizzle(__float_as_int(v), 0x101f)));
    v = fmaxf(v, __int_as_float(__builtin_amdgcn_ds_swizzle(__float_as_int(v), 0x081f)));
    v = fmaxf(v, __int_as_float(__builtin_amdgcn_ds_swizzle(__float_as_int(v), 0x041f)));
    return v;
}

__global__ __launch_bounds__(NTHREADS)
void trans_inv_loss_main(const float* __restrict__ fs_g,   // [T,B,V] first_scores
                         const float* __restrict__ ps_g,   // [T,B,P] pattern_scores
                         const int*   __restrict__ ft_g,   // [T,B]   first_targs
                         const float* __restrict__ pt_g,   // [T,B,P] pattern_targs
                         const int*   __restrict__ len_g,  // [B]     lengths
                         float*       __restrict__ partials) {
    __shared__ int   s_len[B_DIM];
    __shared__ float s_wred[WAVES_PER_BLK];

    // CDNA5 async global->LDS stage of the lengths table (ASYNCcnt-tracked),
    // then workgroup barrier so all waves read it from LDS.
    {
        uint32_t lds_off = (uint32_t)(uintptr_t)(&s_len[threadIdx.x]); // AS(3) offset = low 32 bits
        uint32_t voff    = (uint32_t)threadIdx.x * 4u;
        uint64_t sbase   = (uint64_t)(uintptr_t)len_g;
        asm volatile("global_load_async_to_lds_b32 %0, %1, %2"
                     :: "v"(lds_off), "v"(voff), "s"(sbase)
                     : "memory");
        asm volatile("s_wait_asynccnt 0" ::: "memory");
    }
    __syncthreads();

    const int lane   = threadIdx.x & 31;
    const int wave   = threadIdx.x >> 5;
    const int gwave  = blockIdx.x * WAVES_PER_BLK + wave;
    const int nwaves = gridDim.x * WAVES_PER_BLK;

    float acc = 0.0f;

    for (int row = gwave; row < ROWS; row += nwaves) {
        const int t = row >> 8;            // B_DIM == 256
        const int b = row & (B_DIM - 1);

        // Prefetch next iteration's first_scores row (always consumed):
        // lowers to global_prefetch_b8; 3 lanes cover the 360-byte row.
        const int nrow = row + nwaves;
        if (nrow < ROWS && lane < 3) {
            __builtin_prefetch((const char*)(fs_g + (size_t)nrow * V_DIM) + lane * 128, 0, 0);
        }

        const int targ = ft_g[row];        // wave-uniform -> scalarized load

        // ---------- first-loss: logsumexp over 90 logits, wave-per-row ----------
        const float* fs = fs_g + (size_t)row * V_DIM;
        const float x0 = __builtin_nontemporal_load(fs + lane);
        const float x1 = __builtin_nontemporal_load(fs + lane + 32);
        const float x2 = (lane < V_DIM - 64) ? __builtin_nontemporal_load(fs + lane + 64)
                                             : -INFINITY;

        const float m = wave_max_f32(fmaxf(fmaxf(x0, x1), x2));
        float se = __expf(x0 - m) + __expf(x1 - m);
        if (lane < V_DIM - 64) se += __expf(x2 - m);
        se = wave_sum_f32(se);

        if (targ != IGNORE_IDX) {
            int tc = targ;
            tc = tc < 0 ? 0 : (tc > V_DIM - 1 ? V_DIM - 1 : tc);
            if (lane == 0)       acc += m + __logf(se);  // -logp[targ] = m + log(se) - x[targ]
            if (lane == tc)      acc -= x0;
            if (lane + 32 == tc) acc -= x1;
            if (lane + 64 == tc) acc -= x2;
        }

        // ---------- pattern BCE, prefix-masked: skip masked loads entirely ----------
        int kmax = P_DIM - targ;           // negative when targ==IGNORE
        if (kmax > P_DIM) kmax = P_DIM;
        const int len = s_len[b];
        if (t < len && kmax > 0) {
            const float* ps = ps_g + (size_t)row * P_DIM;
            const float* pt = pt_g + (size_t)row * P_DIM;
#pragma unroll
            for (int r = 0; r < 3; ++r) {
                const int k = lane + r * 32;
                if (k < kmax) {            // kmax <= 89, so k < P_DIM implied
                    const float p = __builtin_nontemporal_load(ps + k);
                    const float y = __builtin_nontemporal_load(pt + k);
                    acc -= y * __logf(p + EPS_F) + (1.0f - y) * __logf(1.0f - p + EPS_F);
                }
            }
        }
    }

    // ---------- deterministic block reduction ----------
    acc = wave_sum_f32(acc);
    if (lane == 0) s_wred[wave] = acc;
    __syncthreads();
    if (threadIdx.x == 0) {
        float tot = 0.0f;
#pragma unroll
        for (int i = 0; i < WAVES_PER_BLK; ++i) tot += s_wred[i];
        partials[blockIdx.x] = tot;
    }
}

__global__ __launch_bounds__(NTHREADS)
void trans_inv_loss_finish(const float* __restrict__ partials, float* __restrict__ out,
                           int nblk) {
    __shared__ float s_wred[WAVES_PER_BLK];
    float a = 0.0f;
    for (int i = threadIdx.x; i < nblk; i += NTHREADS) a += partials[i];
    a = wave_sum_f32(a);
    const int lane = threadIdx.x & 31;
    const int wave = threadIdx.x >> 5;
    if (lane == 0) s_wred[wave] = a;
    __syncthreads();
    if (threadIdx.x == 0) {
        float tot = 0.0f;
#pragma unroll
        for (int i = 0; i < WAVES_PER_BLK; ++i) tot += s_wred[i];
        out[0] = tot * (1.0f / (float)B_DIM);
    }
}

extern "C" void kernel_launch(void* const* d_in, const int* in_sizes, int n_in,
                              void* d_out, int out_size, void* d_ws, size_t ws_size,
                              hipStream_t stream) {
    const float* fs  = (const float*)d_in[0];   // first_scores  [T,B,V] f32
    const float* ps  = (const float*)d_in[1];   // pattern_scores[T,B,P] f32
    const int*   ft  = (const int*)  d_in[2];   // first_targs   [T,B]   i32
    const float* pt  = (const float*)d_in[3];   // pattern_targs [T,B,P] f32
    const int*   len = (const int*)  d_in[4];   // lengths       [B]     i32

    int nblk = MAX_BLOCKS;
    const size_t max_by_ws = ws_size / sizeof(float);
    if ((size_t)nblk > max_by_ws) nblk = (int)max_by_ws;
    if (nblk < 1) nblk = 1;

    float* partials = (float*)d_ws;

    trans_inv_loss_main<<<nblk, NTHREADS, 0, stream>>>(fs, ps, ft, pt, len, partials);
    trans_inv_loss_finish<<<1, NTHREADS, 0, stream>>>(partials, (float*)d_out, nblk);
}